// VIB_61168924230423
// MI455X (gfx1250) — compile-verified
//
#include <hip/hip_runtime.h>
#include <math.h>

#define N_NODES 50000
#define N_EDGES 800000
#define N_PAIRS 100000
#define F_IN    128
#define HID0    128
#define HID1    64

typedef __attribute__((ext_vector_type(2))) float v2f;
typedef __attribute__((ext_vector_type(8))) float v8f;

// ---------------------------------------------------------------- utilities
__global__ void zero_f32(float* __restrict__ p, int n) {
    int i = blockIdx.x * blockDim.x + threadIdx.x;
    if (i < n) p[i] = 0.0f;
}

__global__ void zero_f32x4(float4* __restrict__ p, int n4) {
    int i = blockIdx.x * blockDim.x + threadIdx.x;
    if (i < n4) p[i] = make_float4(0.f, 0.f, 0.f, 0.f);
}

// deg[dst] += 1  (counts only incoming edges, matching segment_sum over dst)
__global__ void deg_count(const int* __restrict__ dst, float* __restrict__ deg) {
    int e = blockIdx.x * blockDim.x + threadIdx.x;   // grid sized exactly
    atomicAdd(&deg[dst[e]], 1.0f);
}

// dis[n] = rsqrt(deg[n] + 1)   (self-loop) — in place
__global__ void deg_rsqrt(float* __restrict__ dis, int n) {
    int i = blockIdx.x * blockDim.x + threadIdx.x;
    if (i < n) dis[i] = rsqrtf(dis[i] + 1.0f);
}

// ------------------------------------------------------- WMMA f32 GEMM
// C[M,N] = A[M,K] * B[K,N],  K == 128, M multiple of 16, N = 16 * wavesPerBlock.
// The 16xK A tile is shared by every wave in the block -> stage it in LDS once
// (padded stride 132 so rows 0..15 land in distinct banks; {K,K+1} fragment is
// one 8B-aligned ds_load_b64). One wave computes one 16x16 output tile via
// V_WMMA_F32_16X16X4_F32.
__global__ void gemm_wmma_f32(const float* __restrict__ A,
                              const float* __restrict__ B,
                              float* __restrict__ C,
                              int N, int K) {
    constexpr int LDA = 132;                 // 128 + 4 pad (banks & 8B align)
    __shared__ float As[16 * LDA];           // 8.25 KB of the 320 KB/WGP LDS

    const int lane = threadIdx.x & 31;
    const int wave = threadIdx.x >> 5;
    const int half = lane >> 4;              // 0: lanes 0-15, 1: lanes 16-31
    const int l16  = lane & 15;
    const int mrow0 = blockIdx.x * 16;       // A/C tile row base

    // cooperative stage: 16*K floats, float4 granularity, uniform trip count
    const int nchunks = 16 * (K / 4);
    for (int idx = threadIdx.x; idx < nchunks; idx += blockDim.x) {
        const int r  = idx / (K / 4);
        const int c4 = idx - r * (K / 4);
        const float4 v = ((const float4*)(A + (size_t)(mrow0 + r) * K))[c4];
        float* dp = &As[r * LDA + c4 * 4];
        dp[0] = v.x; dp[1] = v.y; dp[2] = v.z; dp[3] = v.w;
    }
    __syncthreads();

    const int bcol = wave * 16 + l16;        // B/C column this lane owns
    v8f acc = {};
#pragma unroll 8
    for (int kk = 0; kk < K; kk += 4) {
        const int kb = kk + 2 * half;        // ISA A-layout: K 0,1 / 2,3 per half
        const v2f a = *(const v2f*)&As[l16 * LDA + kb];   // ds_load_b64
        v2f b;
        b.x = B[(size_t)kb * N + bcol];
        b.y = B[(size_t)(kb + 1) * N + bcol];
        acc = __builtin_amdgcn_wmma_f32_16x16x4_f32(
                  false, a, false, b, (short)0, acc, false, false);
    }

    // D layout: VGPR i -> row (i + 8*half), col = lane&15 within the tile
#pragma unroll
    for (int i = 0; i < 8; ++i)
        C[(size_t)(mrow0 + i + 8 * half) * N + bcol] = acc[i];
}

// --------------------------------------------- edge gather + scatter-add
// agg[dst] += h[src] * dis[src]*dis[dst].
// One wave per edge (indices forced scalar via readfirstlane), 4 edges per
// wave with one-edge-lookahead prefetch of the randomly indexed h[src] row.
template <int F>
__global__ void edge_scatter(const int* __restrict__ src,
                             const int* __restrict__ dst,
                             const float* __restrict__ h,
                             const float* __restrict__ dis,
                             float* __restrict__ agg) {
    constexpr int CPL = F / 32;   // floats per lane: 4 (F=128) or 2 (F=64)
    constexpr int EPW = 4;        // edges per wave
    const int lane = threadIdx.x & 31;
    const int wgid = __builtin_amdgcn_readfirstlane(
        blockIdx.x * (blockDim.x >> 5) + (threadIdx.x >> 5));
    const int e0 = wgid * EPW;    // grid sized exactly: e0+EPW <= N_EDGES

    int s = src[e0];
#pragma unroll
    for (int it = 0; it < EPW; ++it) {
        const int e = e0 + it;
        const int d = dst[e];
        const float coef = dis[s] * dis[d];
        const float* hrow = h   + (size_t)s * F + lane * CPL;
        float*       arow = agg + (size_t)d * F + lane * CPL;
        if (it + 1 < EPW) {       // software pipeline: prefetch next gather row
            const int sn = src[e + 1];
            __builtin_prefetch(h + (size_t)sn * F + lane * CPL, 0, 0);
            s = sn;
        }
        if constexpr (CPL == 4) {
            const float4 hv = *(const float4*)hrow;
            atomicAdd(arow + 0, hv.x * coef);
            atomicAdd(arow + 1, hv.y * coef);
            atomicAdd(arow + 2, hv.z * coef);
            atomicAdd(arow + 3, hv.w * coef);
        } else {
            const float2 hv = *(const float2*)hrow;
            atomicAdd(arow + 0, hv.x * coef);
            atomicAdd(arow + 1, hv.y * coef);
        }
    }
}

// out = (agg + h*dis^2 + bias)  [+ ReLU]   — safe in place (out == agg)
template <int F, bool RELU>
__global__ void finalize_nodes(const float* agg,
                               const float* __restrict__ h,
                               const float* __restrict__ dis,
                               const float* __restrict__ bias,
                               float* out) {
    const int t = blockIdx.x * blockDim.x + threadIdx.x;  // N_NODES*F threads
    const int n = t / F;
    const int f = t - n * F;
    const float di = dis[n];
    float v = agg[t] + h[t] * (di * di) + bias[f];
    if (RELU) v = fmaxf(v, 0.0f);
    out[t] = v;
}

// ----------------------------------------------------------- VIB pair head
__global__ void pair_head(const float* __restrict__ emb,
                          const int* __restrict__ x1,
                          const int* __restrict__ x2,
                          const float* __restrict__ log_a,
                          const float* __restrict__ log_a_f,
                          const float* __restrict__ bb,
                          const float* __restrict__ bf,
                          float* __restrict__ out) {
    const int p = blockIdx.x * blockDim.x + threadIdx.x;
    if (p >= N_PAIRS) return;
    const float4* e1 = (const float4*)(emb + (size_t)x1[p] * HID1);
    const float4* e2 = (const float4*)(emb + (size_t)x2[p] * HID1);
    float dist = 0.0f;
#pragma unroll
    for (int i = 0; i < HID1 / 4; ++i) {
        const float4 a = e1[i], b = e2[i];
        const float dx = a.x - b.x, dy = a.y - b.y;
        const float dz = a.z - b.z, dw = a.w - b.w;
        dist += dx * dx + dy * dy + dz * dz + dw * dw;
    }
    const float ga = expf(log_a[0]);
    const float gf = expf(log_a_f[0]);
    // sigmoid(-a*d + b) = 1 / (1 + exp(a*d - b))
    out[p]           = 1.0f / (1.0f + expf(ga * dist - bb[0]));
    out[N_PAIRS + p] = 1.0f / (1.0f + expf(gf * dist - bf[0]));
}

// ------------------------------------------------------------------ driver
extern "C" void kernel_launch(void* const* d_in, const int* in_sizes, int n_in,
                              void* d_out, int out_size, void* d_ws, size_t ws_size,
                              hipStream_t stream) {
    const float* graph  = (const float*)d_in[0];
    const int*   edge   = (const int*)  d_in[1];
    const int*   x1     = (const int*)  d_in[2];
    const int*   x2     = (const int*)  d_in[3];
    const float* W1     = (const float*)d_in[4];
    const float* b1     = (const float*)d_in[5];
    const float* W2     = (const float*)d_in[6];
    const float* b2     = (const float*)d_in[7];
    const float* log_a  = (const float*)d_in[8];
    const float* log_af = (const float*)d_in[9];
    const float* bb     = (const float*)d_in[10];
    const float* bf     = (const float*)d_in[11];

    const int* esrc = edge;            // edge_index[0, :]
    const int* edst = edge + N_EDGES;  // edge_index[1, :]

    // workspace layout (floats); buffers from layer 1 recycled for layer 2
    float* ws_f = (float*)d_ws;
    float* dis  = ws_f;                        // 50000   (padded to 50176)
    float* Hbuf = ws_f + 50176;                // 50000*128 = 6,400,000
    float* AGG1 = Hbuf + 6400000;              // 6,400,000 (becomes H1 in place)
    float* H2   = Hbuf;                        // 3,200,000 (H dead after layer 1)
    float* AGG2 = Hbuf + 3200000;              // 3,200,000 (becomes emb in place)
    float* outp = (float*)d_out;

    // ---- degrees ----
    zero_f32<<<(N_NODES + 255) / 256, 256, 0, stream>>>(dis, N_NODES);
    deg_count<<<N_EDGES / 256, 256, 0, stream>>>(edst, dis);
    deg_rsqrt<<<(N_NODES + 255) / 256, 256, 0, stream>>>(dis, N_NODES);

    // ---- layer 1:  H = X @ W1 ----
    gemm_wmma_f32<<<N_NODES / 16, 256, 0, stream>>>(graph, W1, Hbuf, HID0, F_IN);
    zero_f32x4<<<(N_NODES * HID0 / 4) / 256, 256, 0, stream>>>((float4*)AGG1,
                                                               N_NODES * HID0 / 4);
    edge_scatter<HID0><<<N_EDGES / (4 * 8), 256, 0, stream>>>(
        esrc, edst, Hbuf, dis, AGG1);
    finalize_nodes<HID0, true><<<(N_NODES * HID0) / 256, 256, 0, stream>>>(
        AGG1, Hbuf, dis, b1, AGG1);                      // AGG1 -> H1 (ReLU'd)

    // ---- layer 2:  H2 = H1 @ W2 ----
    gemm_wmma_f32<<<N_NODES / 16, 128, 0, stream>>>(AGG1, W2, H2, HID1, HID0);
    zero_f32x4<<<(N_NODES * HID1 / 4) / 256, 256, 0, stream>>>((float4*)AGG2,
                                                               N_NODES * HID1 / 4);
    edge_scatter<HID1><<<N_EDGES / (4 * 8), 256, 0, stream>>>(
        esrc, edst, H2, dis, AGG2);
    finalize_nodes<HID1, false><<<(N_NODES * HID1) / 256, 256, 0, stream>>>(
        AGG2, H2, dis, b2, AGG2);                        // AGG2 -> emb

    // ---- VIB pair head ----
    pair_head<<<(N_PAIRS + 255) / 256, 256, 0, stream>>>(
        AGG2, x1, x2, log_a, log_af, bb, bf, outp);
}